// GATNEModel_29094108463174
// MI455X (gfx1250) — compile-verified
//
#include <hip/hip_runtime.h>
#include <hip/hip_bf16.h>

typedef __attribute__((ext_vector_type(2))) float v2f;
typedef __attribute__((ext_vector_type(8))) float v8f;

#define TT 8     // edge types
#define NNB 10   // neighbors per type
#define FF 256   // feature dim
#define EE 256   // embed dim
#define UU 32    // u-embed dim
#define AA 32    // attention dim

// ---------------------------------------------------------------------------
// Kernel 1: base embedding  out[b,:] = features[train_inputs[b],:] @ embed_trans
// block = 128 threads (4 waves), each block does a 16-row stripe of B.
// A tile (16x256 f32) staged in LDS; each wave computes 4 x (16x16) tiles of E
// with a K-loop of v_wmma_f32_16x16x4_f32.
// ---------------------------------------------------------------------------
__global__ __launch_bounds__(128)
void k_base_embed(const int* __restrict__ train_inputs,
                  const float* __restrict__ features,
                  const float* __restrict__ embed_trans,
                  float* __restrict__ out)
{
    __shared__ float As[16][FF];
    const int bbase = blockIdx.x * 16;
    const int tid   = threadIdx.x;

    // cooperative gather: each thread loads 32 floats (8 x float4)
    {
        const int r  = tid >> 3;             // 0..15
        const int c0 = (tid & 7) * 32;       // 0,32,...,224
        const long row = (long)train_inputs[bbase + r] * FF;
        #pragma unroll
        for (int j = 0; j < 8; ++j) {
            const float4 v = *reinterpret_cast<const float4*>(&features[row + c0 + j * 4]);
            *reinterpret_cast<float4*>(&As[r][c0 + j * 4]) = v;
        }
    }
    __syncthreads();

    const int wave  = tid >> 5;
    const int lane  = tid & 31;
    const int mrow  = lane & 15;      // M row for A fragment
    const int khalf = lane >> 4;      // K half select
    const int ncol  = lane & 15;      // N column for B/D fragments
    const int n0    = wave * 64;      // this wave covers columns [n0, n0+64)

    v8f acc0 = {}, acc1 = {}, acc2 = {}, acc3 = {};
    for (int k0 = 0; k0 < FF; k0 += 4) {
        const int rk = k0 + 2 * khalf;
        // A fragment: lane holds A[mrow][rk], A[mrow][rk+1]
        const float2 av = *reinterpret_cast<const float2*>(&As[mrow][rk]);
        v2f a; a.x = av.x; a.y = av.y;
        // B fragments: VGPR0 = B[rk][n], VGPR1 = B[rk+1][n]
        const float* B0 = &embed_trans[(long)rk * EE + n0 + ncol];
        v2f b;
        b.x = B0[0];       b.y = B0[EE];
        acc0 = __builtin_amdgcn_wmma_f32_16x16x4_f32(false, a, false, b, (short)0, acc0, false, false);
        b.x = B0[16];      b.y = B0[EE + 16];
        acc1 = __builtin_amdgcn_wmma_f32_16x16x4_f32(false, a, false, b, (short)0, acc1, false, false);
        b.x = B0[32];      b.y = B0[EE + 32];
        acc2 = __builtin_amdgcn_wmma_f32_16x16x4_f32(false, a, false, b, (short)0, acc2, false, false);
        b.x = B0[48];      b.y = B0[EE + 48];
        acc3 = __builtin_amdgcn_wmma_f32_16x16x4_f32(false, a, false, b, (short)0, acc3, false, false);
    }

    // D layout: VGPR r -> row (r + 8*khalf), col = ncol
    #pragma unroll
    for (int r = 0; r < 8; ++r) {
        float* o = &out[(long)(bbase + r + 8 * khalf) * EE + n0 + ncol];
        o[0]  = acc0[r];
        o[16] = acc1[r];
        o[32] = acc2[r];
        o[48] = acc3[r];
    }
}

// ---------------------------------------------------------------------------
// Kernel 2: nte[b,t,:] = (sum_n features[node_neigh[b,t,n],:]) @ u_embed_trans[t]
// block = 128 threads; grid = (B/16, T). Gather-sum into LDS, then waves 0,1
// each produce one 16x16 tile of the 16x32 result via WMMA.
// ---------------------------------------------------------------------------
__global__ __launch_bounds__(128)
void k_type_embed(const int* __restrict__ node_neigh,
                  const float* __restrict__ features,
                  const float* __restrict__ u_embed_trans,
                  float* __restrict__ nte)            // [B][T][UU]
{
    __shared__ float As[16][FF];
    const int t     = blockIdx.y;
    const int bbase = blockIdx.x * 16;
    const int tid   = threadIdx.x;

    // gather + sum over 10 neighbors: each thread owns 32 columns of one row
    {
        const int r  = tid >> 3;
        const int c0 = (tid & 7) * 32;
        const int* nb = &node_neigh[((long)(bbase + r) * TT + t) * NNB];
        float4 acc[8];
        #pragma unroll
        for (int j = 0; j < 8; ++j) acc[j] = make_float4(0.f, 0.f, 0.f, 0.f);
        for (int nn = 0; nn < NNB; ++nn) {
            const long row = (long)nb[nn] * FF;
            #pragma unroll
            for (int j = 0; j < 8; ++j) {
                const float4 v = *reinterpret_cast<const float4*>(&features[row + c0 + j * 4]);
                acc[j].x += v.x; acc[j].y += v.y; acc[j].z += v.z; acc[j].w += v.w;
            }
        }
        #pragma unroll
        for (int j = 0; j < 8; ++j)
            *reinterpret_cast<float4*>(&As[r][c0 + j * 4]) = acc[j];
    }
    __syncthreads();

    const int wave = tid >> 5;
    if (wave >= 2) return;            // wave-uniform: EXEC stays all-ones below
    const int lane  = tid & 31;
    const int mrow  = lane & 15;
    const int khalf = lane >> 4;
    const int ncol  = lane & 15;
    const int n0    = wave * 16;
    const float* Um = &u_embed_trans[(long)t * FF * UU];

    v8f acc = {};
    for (int k0 = 0; k0 < FF; k0 += 4) {
        const int rk = k0 + 2 * khalf;
        const float2 av = *reinterpret_cast<const float2*>(&As[mrow][rk]);
        v2f a; a.x = av.x; a.y = av.y;
        const float* B0 = &Um[(long)rk * UU + n0 + ncol];
        v2f b; b.x = B0[0]; b.y = B0[UU];
        acc = __builtin_amdgcn_wmma_f32_16x16x4_f32(false, a, false, b, (short)0, acc, false, false);
    }
    #pragma unroll
    for (int r = 0; r < 8; ++r)
        nte[((long)(bbase + r + 8 * khalf) * TT + t) * UU + n0 + ncol] = acc[r];
}

// ---------------------------------------------------------------------------
// Kernel 3: attention over edge types + pooled transform + row normalize.
// One wave32 per batch row; 8 waves per block.
// ---------------------------------------------------------------------------
__global__ __launch_bounds__(256)
void k_attn_finalize(const int* __restrict__ train_types,
                     const float* __restrict__ trans_weights,     // [T][UU][EE]
                     const float* __restrict__ trans_weights_s1,  // [T][UU][AA]
                     const float* __restrict__ trans_weights_s2,  // [T][AA]
                     const float* __restrict__ nte,               // [B][T][UU]
                     float* __restrict__ out, int B)
{
    const int wave = threadIdx.x >> 5;
    const int lane = threadIdx.x & 31;
    const int b = blockIdx.x * 8 + wave;
    if (b >= B) return;

    const int type  = train_types[b];
    const float* S1 = &trans_weights_s1[(long)type * UU * AA];
    const float s2l = trans_weights_s2[(long)type * AA + lane];

    // lane u holds nte[b][t][u]
    float val[TT];
    #pragma unroll
    for (int t = 0; t < TT; ++t)
        val[t] = nte[((long)b * TT + t) * UU + lane];

    // scores[t] = sum_a tanh(sum_u val[t][u]*S1[u][a]) * S2[a]
    float score[TT];
    #pragma unroll
    for (int t = 0; t < TT; ++t) {
        float h = 0.f;
        for (int u = 0; u < UU; ++u) {
            const float vu = __shfl(val[t], u, 32);
            h += vu * S1[u * AA + lane];
        }
        float c = tanhf(h) * s2l;
        #pragma unroll
        for (int off = 16; off > 0; off >>= 1)
            c += __shfl_xor(c, off, 32);   // all lanes get the sum
        score[t] = c;
    }

    // softmax over T, pooled[u] = sum_t att[t]*val[t][u]   (lane u owns pooled[u])
    float mx = score[0];
    #pragma unroll
    for (int t = 1; t < TT; ++t) mx = fmaxf(mx, score[t]);
    float den = 0.f, att[TT];
    #pragma unroll
    for (int t = 0; t < TT; ++t) { att[t] = __expf(score[t] - mx); den += att[t]; }
    const float inv = 1.0f / den;
    float pooled = 0.f;
    #pragma unroll
    for (int t = 0; t < TT; ++t) pooled += att[t] * inv * val[t];

    // out[b,:] += pooled @ trans_weights[type]  (lane covers cols lane+32*j)
    const float* W = &trans_weights[(long)type * UU * EE];
    float acc[8];
    #pragma unroll
    for (int j = 0; j < 8; ++j) acc[j] = out[(long)b * EE + lane + 32 * j];
    for (int u = 0; u < UU; ++u) {
        const float p = __shfl(pooled, u, 32);
        const float* Wr = &W[(long)u * EE + lane];
        #pragma unroll
        for (int j = 0; j < 8; ++j) acc[j] += p * Wr[32 * j];
    }

    // L2 normalize the row
    float ss = 0.f;
    #pragma unroll
    for (int j = 0; j < 8; ++j) ss += acc[j] * acc[j];
    #pragma unroll
    for (int off = 16; off > 0; off >>= 1)
        ss += __shfl_xor(ss, off, 32);
    const float scale = 1.0f / fmaxf(sqrtf(ss), 1e-12f);
    #pragma unroll
    for (int j = 0; j < 8; ++j)
        out[(long)b * EE + lane + 32 * j] = acc[j] * scale;
}

// ---------------------------------------------------------------------------
extern "C" void kernel_launch(void* const* d_in, const int* in_sizes, int n_in,
                              void* d_out, int out_size, void* d_ws, size_t ws_size,
                              hipStream_t stream)
{
    (void)n_in; (void)out_size; (void)ws_size;
    const int*   train_inputs  = (const int*)d_in[0];
    const int*   train_types   = (const int*)d_in[1];
    const int*   node_neigh    = (const int*)d_in[2];
    const float* features      = (const float*)d_in[3];
    const float* embed_trans   = (const float*)d_in[4];
    const float* u_embed_trans = (const float*)d_in[5];
    const float* trans_weights = (const float*)d_in[6];
    const float* tw_s1         = (const float*)d_in[7];
    const float* tw_s2         = (const float*)d_in[8];
    float* out = (float*)d_out;
    float* nte = (float*)d_ws;              // [B][T][UU] scratch = B*8*32*4 bytes

    const int B = in_sizes[0];

    k_base_embed<<<B / 16, 128, 0, stream>>>(train_inputs, features, embed_trans, out);

    dim3 g2(B / 16, TT);
    k_type_embed<<<g2, 128, 0, stream>>>(node_neigh, features, u_embed_trans, nte);

    k_attn_finalize<<<B / 8, 256, 0, stream>>>(train_types, trans_weights, tw_s1, tw_s2,
                                               nte, out, B);
}